// Caps_Layer_70308614635891
// MI455X (gfx1250) — compile-verified
//
#include <hip/hip_runtime.h>
#include <hip/hip_bf16.h>

typedef __attribute__((ext_vector_type(2))) float v2f;
typedef __attribute__((ext_vector_type(8))) float v8f;

#define NTOK 512
#define DDIM 300
#define NCAP 10
#define DCAP 64
#define MDIM 640   // NCAP*DCAP

// ---------------------------------------------------------------------------
// GEMM: u_hat[b][i][n][k] = sum_d x[b][n][d] * W[d][i*64+k]
// One block per (m-tile of 16 tokens, batch). 8 waves, each wave computes a
// 16x80 strip of columns using 5 f32 WMMA accumulators. A tile staged in LDS.
// ---------------------------------------------------------------------------
__global__ __launch_bounds__(256) void caps_gemm_wmma(
    const float* __restrict__ x,   // [128][512][300]
    const float* __restrict__ W,   // [300][640]
    float* __restrict__ u)         // [128][10][512][64]
{
    __shared__ float xs[16 * DDIM];          // 19.2 KB A tile

    const int b    = blockIdx.z;
    const int m0   = blockIdx.x * 16;        // token base
    const int wave = threadIdx.x >> 5;       // 0..7
    const int lane = threadIdx.x & 31;
    const int half = lane >> 4;              // K-half select (ISA A layout)
    const int l15  = lane & 15;

    // cooperative load of 16x300 x-tile (coalesced, 4800 floats / 256 thr)
    const float* xsrc = x + ((size_t)b * NTOK + m0) * DDIM;
    for (int idx = threadIdx.x; idx < 16 * DDIM; idx += 256)
        xs[idx] = xsrc[idx];
    __syncthreads();

    const float* arow = xs + l15 * DDIM;     // row M = lane&15 (conflict-free)
    v8f acc[5] = {};

    for (int k0 = 0; k0 < DDIM; k0 += 4) {
        // A fragment: lanes 0-15 hold K = k0+{0,1}; lanes 16-31 K = k0+{2,3}
        v2f a;
        a.x = arow[k0 + 2 * half + 0];
        a.y = arow[k0 + 2 * half + 1];
        // B fragment source: rows K of W, column (n0 + lane&15)
        const float* wr = W + (size_t)(k0 + 2 * half) * MDIM + l15;
#pragma unroll
        for (int q = 0; q < 5; ++q) {
            const int n0 = wave * 80 + q * 16;
            v2f bm;
            bm.x = wr[n0];
            bm.y = wr[MDIM + n0];
            acc[q] = __builtin_amdgcn_wmma_f32_16x16x4_f32(
                /*neg_a=*/false, a, /*neg_b=*/false, bm,
                /*c_mod=*/(short)0, acc[q],
                /*reuse_a=*/false, /*reuse_b=*/false);
        }
    }

    // Scatter D (VGPR r: lanes0-15 -> M=r, lanes16-31 -> M=r+8; N = lane&15)
    // into [b][cap][token][dim] layout for the routing phase.
#pragma unroll
    for (int q = 0; q < 5; ++q) {
        const int n0    = wave * 80 + q * 16;
        const int cap   = n0 >> 6;           // 16 | 64, tile never crosses cap
        const int kbase = (n0 & 63) + l15;
        float* uo = u + (((size_t)b * NCAP + cap) * NTOK) * DCAP + kbase;
#pragma unroll
        for (int r = 0; r < 8; ++r) {
            const int M = r + half * 8;
            uo[(size_t)(m0 + M) * DCAP] = acc[q][r];
        }
    }
}

// ---------------------------------------------------------------------------
// Dynamic routing, 5 iterations fused. One 640-thread block per batch element.
// Logits L[10][512], coupling C[10][512], outputs O[10][64] all in LDS.
// ---------------------------------------------------------------------------
__global__ __launch_bounds__(640) void caps_route(
    const float* __restrict__ u,   // [128][10][512][64]
    float* __restrict__ out)       // [128][10][64]
{
    __shared__ float L[NCAP][NTOK];
    __shared__ float C[NCAP][NTOK];
    __shared__ float O[NCAP][DCAP];
    __shared__ float Sinv[NCAP];

    const int b   = blockIdx.x;
    const int tid = threadIdx.x;
    const int ci  = tid >> 6;                // capsule 0..9
    const int ck  = tid & 63;                // dim 0..63
    const float* ub = u + (size_t)b * NCAP * NTOK * DCAP;

    for (int idx = tid; idx < NCAP * NTOK; idx += 640)
        (&L[0][0])[idx] = 0.0f;
    __syncthreads();

    for (int it = 0; it < 5; ++it) {
        // c = softmax over capsule axis, per token j
        for (int j = tid; j < NTOK; j += 640) {
            float mx = L[0][j];
#pragma unroll
            for (int i = 1; i < NCAP; ++i) mx = fmaxf(mx, L[i][j]);
            float e[NCAP], s = 0.0f;
#pragma unroll
            for (int i = 0; i < NCAP; ++i) { e[i] = __expf(L[i][j] - mx); s += e[i]; }
            const float inv = 1.0f / s;
#pragma unroll
            for (int i = 0; i < NCAP; ++i) C[i][j] = e[i] * inv;
        }
        __syncthreads();

        // O[i][k] = sum_j C[i][j] * u[i][j][k]  (coalesced over k)
        {
            const float* up = ub + ((size_t)ci * NTOK) * DCAP + ck;
            float s = 0.0f;
            for (int j = 0; j < NTOK; ++j)
                s = fmaf(C[ci][j], up[(size_t)j * DCAP], s);
            O[ci][ck] = s;
        }
        __syncthreads();

        // squash: O[i][:] /= sqrt(sum_k O^2 + eps)
        if (tid < NCAP) {
            float s = 0.0f;
#pragma unroll
            for (int k = 0; k < DCAP; ++k) s = fmaf(O[tid][k], O[tid][k], s);
            Sinv[tid] = rsqrtf(s + 1e-7f);
        }
        __syncthreads();
        O[ci][ck] *= Sinv[ci];
        __syncthreads();

        if (it < 4) {
            // L[i][j] = sum_k O[i][k] * u[i][j][k]; one wave per (i,j) pair,
            // lanes split k (coalesced 256B rows), wave32 shuffle reduction.
            const int wv = tid >> 5, ln = tid & 31;      // 20 waves
            for (int p = wv; p < NCAP * NTOK; p += 20) {
                const int i = p >> 9, j = p & (NTOK - 1);
                const float* up = ub + (((size_t)i * NTOK) + j) * DCAP;
                float s = fmaf(O[i][ln], up[ln], O[i][ln + 32] * up[ln + 32]);
#pragma unroll
                for (int off = 16; off >= 1; off >>= 1)
                    s += __shfl_xor(s, off, 32);
                if (ln == 0) L[i][j] = s;
            }
            __syncthreads();
        }
    }

    out[((size_t)b * NCAP + ci) * DCAP + ck] = O[ci][ck];
}

extern "C" void kernel_launch(void* const* d_in, const int* in_sizes, int n_in,
                              void* d_out, int out_size, void* d_ws, size_t ws_size,
                              hipStream_t stream) {
    (void)in_sizes; (void)n_in; (void)out_size; (void)ws_size;
    const float* x = (const float*)d_in[0];   // [128,512,300] f32
    const float* W = (const float*)d_in[1];   // [1,300,640]   f32
    float* out = (float*)d_out;               // [128,10,64]   f32
    float* u   = (float*)d_ws;                // u_hat: 128*10*512*64 f32 (167.8 MB)

    dim3 g1(NTOK / 16, 1, 128);               // 32 m-tiles x 128 batches
    caps_gemm_wmma<<<g1, 256, 0, stream>>>(x, W, u);
    caps_route<<<128, 640, 0, stream>>>(u, out);
}